// CrossAttention_46918222742033
// MI455X (gfx1250) — compile-verified
//
#include <hip/hip_runtime.h>
#include <hip/hip_bf16.h>

typedef __attribute__((ext_vector_type(2))) float v2f;
typedef __attribute__((ext_vector_type(8))) float v8f;

#define BB 4
#define CC 64
#define KK 64
#define DD 16
#define HH 64
#define WW 64
#define SQ 65536                       // D*H*W
#define CHAN_ELEMS ((size_t)CC * SQ)   // per-batch q/k/v elements

// ---------------------------------------------------------------------------
// WMMA f32 16x16x4 (codegen-confirmed in round 1: lowers to
// v_wmma_f32_16x16x4_f32).
// ---------------------------------------------------------------------------
#if __has_builtin(__builtin_amdgcn_wmma_f32_16x16x4_f32)
__device__ __forceinline__ v8f wmma_f32x4(v2f a, v2f b, v8f c) {
  return __builtin_amdgcn_wmma_f32_16x16x4_f32(false, a, false, b, (short)0, c,
                                               false, false);
}
#else
__device__ __forceinline__ v8f wmma_f32x4(v2f a, v2f b, v8f c) {
  c[0] += a.x * b.x + a.y * b.y;
  return c;
}
#endif

// ---------------------------------------------------------------------------
// Async memory->LDS copy engine (ASYNCcnt-tracked, no VGPR round-trip).
// GV mode: each lane copies 16B from global VADDR to the LDS offset in VDST. LDS byte offset = low 32 bits of
// the generic shared pointer (aperture occupies addr[63:32] only).
// ---------------------------------------------------------------------------
__device__ __forceinline__ void async_load_b128_to_lds(const float* lds_dst,
                                                       const float* gsrc) {
  const unsigned int lds_off = (unsigned int)(unsigned long long)lds_dst;
  asm volatile("global_load_async_to_lds_b128 %0, %1, off" ::"v"(lds_off),
               "v"(gsrc)
               : "memory");
}

__device__ __forceinline__ void wait_asynccnt0() {
#if __has_builtin(__builtin_amdgcn_s_wait_asynccnt)
  __builtin_amdgcn_s_wait_asynccnt(0);
#else
  asm volatile("s_wait_asynccnt 0x0" ::: "memory");
#endif
}

// ---------------------------------------------------------------------------
// Kernel 1: depthwise 3x3x3 conv, stride 1, SAME. One thread per output elem,
// w fastest -> coalesced streaming; taps hit WGP$/L2.
// ---------------------------------------------------------------------------
__global__ __launch_bounds__(256) void dwconv3d_kernel(
    const float* __restrict__ x, const float* __restrict__ w,
    const float* __restrict__ bias, float* __restrict__ y, int total) {
  int idx = blockIdx.x * blockDim.x + threadIdx.x;
  if (idx >= total) return;
  const int wi = idx & 63;
  const int hi = (idx >> 6) & 63;
  const int di = (idx >> 12) & 15;
  const int c = (idx >> 16) & 63;
  const float* __restrict__ xc = x + ((size_t)(idx >> 16) << 16);  // (b,c)
  const float* __restrict__ wc = w + c * 27;

  float sum = bias[c];
#pragma unroll
  for (int kd = -1; kd <= 1; ++kd) {
    const int dd = di + kd;
    if (dd < 0 || dd >= DD) continue;
#pragma unroll
    for (int kh = -1; kh <= 1; ++kh) {
      const int hh = hi + kh;
      if (hh < 0 || hh >= HH) continue;
#pragma unroll
      for (int kw = -1; kw <= 1; ++kw) {
        const int ww = wi + kw;
        if (ww < 0 || ww >= WW) continue;
        sum = fmaf(xc[dd * 4096 + hh * 64 + ww],
                   wc[(kd + 1) * 9 + (kh + 1) * 3 + (kw + 1)], sum);
      }
    }
  }
  y[idx] = sum;
}

// ---------------------------------------------------------------------------
// Kernel 2: attn[b,c,k] = sum_s q[b,c,s]*k[b,k,s]  (split-K + f32 WMMA)
// 512 threads = 16 waves; the workgroup async-copies full 64-row q/k s-slices
// into LDS (padded row stride 68 floats: 16B-aligned rows, conflict-free b64
// fragment reads), each wave owns one distinct 16x16 output tile and
// atomically accumulates its split-K partial.
// ---------------------------------------------------------------------------
#define SW 64  // staged s width
#define RS 68  // padded LDS row stride (floats)

__global__ __launch_bounds__(512) void attn_qk_kernel(
    const float* __restrict__ q, const float* __restrict__ kbuf,
    float* __restrict__ attn, int sPerBlock) {
  __shared__ float ldsQ[64 * RS];
  __shared__ float ldsK[64 * RS];

  const int tid = threadIdx.x;
  const int wave = tid >> 5;  // 0..15
  const int lane = tid & 31;
  const int lrow = lane & 15;
  const int halfSel = (lane < 16) ? 0 : 2;  // K-pair select within WMMA step
  const int c0 = (wave >> 2) * 16;
  const int k0 = (wave & 3) * 16;
  const int b = blockIdx.y;
  const int sBase = blockIdx.x * sPerBlock;

  const float* __restrict__ qb = q + (size_t)b * CHAN_ELEMS;
  const float* __restrict__ kb = kbuf + (size_t)b * CHAN_ELEMS;

  v8f acc = {};

  for (int s = sBase; s < sBase + sPerBlock; s += SW) {
    // Stage 64x64 floats of q and k via the async copy engine:
    // 1024 b128 slots per matrix, 512 threads x2 slots.
#pragma unroll
    for (int it = 0; it < 2; ++it) {
      const int slot = tid + it * 512;
      const int row = slot >> 4;
      const int c4 = (slot & 15) << 2;
      async_load_b128_to_lds(ldsQ + row * RS + c4,
                             qb + (size_t)row * SQ + s + c4);
      async_load_b128_to_lds(ldsK + row * RS + c4,
                             kb + (size_t)row * SQ + s + c4);
    }
    wait_asynccnt0();   // this wave's copies are LDS-resident
    __syncthreads();    // ... and therefore everyone's are after the barrier
#pragma unroll
    for (int so = 0; so < SW; so += 4) {
      const v2f a = *(const v2f*)(ldsQ + (c0 + lrow) * RS + so + halfSel);
      const v2f bf = *(const v2f*)(ldsK + (k0 + lrow) * RS + so + halfSel);
      acc = wmma_f32x4(a, bf, acc);
    }
    __syncthreads();
  }

  // D layout: vgpr i -> M = i (+8 for upper half-wave), N = lane&15.
  float* __restrict__ ab = attn + b * (CC * KK);
  const int n = k0 + lrow;
  const int mBase = c0 + ((lane < 16) ? 0 : 8);
#pragma unroll
  for (int i = 0; i < 8; ++i) {
    atomicAdd(ab + (mBase + i) * KK + n, acc[i]);
  }
}

// ---------------------------------------------------------------------------
// Kernel 3: row softmax over K=64 with 1/sqrt(K)=0.125 scale. One wave/row.
// ---------------------------------------------------------------------------
__global__ __launch_bounds__(256) void softmax_kernel(float* __restrict__ attn) {
  const int row = blockIdx.x * 8 + (threadIdx.x >> 5);  // 0..B*C-1
  const int lane = threadIdx.x & 31;
  float* __restrict__ p = attn + row * KK;
  float a0 = p[lane] * 0.125f;
  float a1 = p[lane + 32] * 0.125f;
  float m = fmaxf(a0, a1);
#pragma unroll
  for (int off = 16; off; off >>= 1) m = fmaxf(m, __shfl_xor(m, off, 32));
  const float e0 = __expf(a0 - m);
  const float e1 = __expf(a1 - m);
  float sum = e0 + e1;
#pragma unroll
  for (int off = 16; off; off >>= 1) sum += __shfl_xor(sum, off, 32);
  const float inv = 1.0f / sum;
  p[lane] = e0 * inv;
  p[lane + 32] = e1 * inv;
}

// ---------------------------------------------------------------------------
// Kernel 4: per (b,k) column, keep top-6 along C (ties -> lowest c, matching
// lax.top_k), zero the rest. One wave per column; 6 wave-argmax rounds.
// ---------------------------------------------------------------------------
__global__ __launch_bounds__(256) void topk_mask_kernel(
    const float* __restrict__ attn, float* __restrict__ attn_m) {
  const int col = blockIdx.x * 8 + (threadIdx.x >> 5);  // 0..B*K-1
  const int lane = threadIdx.x & 31;
  const int b = col >> 6;
  const int k = col & 63;
  const float* __restrict__ base = attn + b * (CC * KK);
  const float v0 = base[lane * KK + k];
  const float v1 = base[(lane + 32) * KK + k];
  bool s0 = false, s1 = false;
#pragma unroll
  for (int it = 0; it < 6; ++it) {
    const float c0v = s0 ? -1.0f : v0;  // probs >= 0, -1 == excluded
    const float c1v = s1 ? -1.0f : v1;
    float bv;
    int bi;
    if (c0v >= c1v) { bv = c0v; bi = lane; }  // lower index wins ties
    else            { bv = c1v; bi = lane + 32; }
#pragma unroll
    for (int off = 16; off; off >>= 1) {
      const float ov = __shfl_xor(bv, off, 32);
      const int oi = __shfl_xor(bi, off, 32);
      if (ov > bv || (ov == bv && oi < bi)) { bv = ov; bi = oi; }
    }
    if (bi == lane) s0 = true;
    else if (bi == lane + 32) s1 = true;
  }
  float* __restrict__ om = attn_m + b * (CC * KK);
  om[lane * KK + k] = s0 ? v0 : 0.0f;
  om[(lane + 32) * KK + k] = s1 ? v1 : 0.0f;
}

// ---------------------------------------------------------------------------
// Kernel 5: out[b,c,s] = x_q[b,c,s] + sum_k attn_m[b,c,k] * v[b,k,s]
// A-fragments (attn_m rows) preloaded to 16 v2f registers; B streamed from v
// with 64B-coalesced lane loads; residual add fused into D writeback.
// ---------------------------------------------------------------------------
__global__ __launch_bounds__(256) void attn_v_kernel(
    const float* __restrict__ attn_m, const float* __restrict__ vbuf,
    const float* __restrict__ xq, float* __restrict__ out) {
  const int lane = threadIdx.x & 31;
  const int wave = threadIdx.x >> 5;  // 0..7
  const int lrow = lane & 15;
  const int halfSel = (lane < 16) ? 0 : 2;
  const int b = blockIdx.z;
  const int c0 = blockIdx.y * 16;
  const int sBase = (blockIdx.x * 8 + wave) * 256;  // 16 s-tiles per wave

  const float* __restrict__ am = attn_m + b * (CC * KK);
  const float* __restrict__ vb = vbuf + (size_t)b * CHAN_ELEMS;
  const float* __restrict__ xb = xq + (size_t)b * CHAN_ELEMS;
  float* __restrict__ ob = out + (size_t)b * CHAN_ELEMS;

  v2f afr[16];
#pragma unroll
  for (int kk = 0; kk < 16; ++kk) {
    afr[kk] = *(const v2f*)(am + (c0 + lrow) * KK + kk * 4 + halfSel);
  }

  for (int st = 0; st < 16; ++st) {
    const int s0 = sBase + st * 16;
    v8f acc = {};
#pragma unroll
    for (int kk = 0; kk < 16; ++kk) {
      const int krow = kk * 4 + halfSel;
      v2f bf;
      bf.x = vb[(size_t)krow * SQ + s0 + lrow];
      bf.y = vb[(size_t)(krow + 1) * SQ + s0 + lrow];
      acc = wmma_f32x4(afr[kk], bf, acc);
    }
    const int n = s0 + lrow;
    const int mBase = c0 + ((lane < 16) ? 0 : 8);
#pragma unroll
    for (int i = 0; i < 8; ++i) {
      const size_t off = (size_t)(mBase + i) * SQ + n;
      ob[off] = xb[off] + acc[i];
    }
  }
}

// ---------------------------------------------------------------------------
// Launch: ws layout = q | k | v | attn  (3*64MB + 64KB)
// d_out = [B,C,D,H,W] residual output  ++  [B,C,K] attn_m
// ---------------------------------------------------------------------------
extern "C" void kernel_launch(void* const* d_in, const int* in_sizes, int n_in,
                              void* d_out, int out_size, void* d_ws,
                              size_t ws_size, hipStream_t stream) {
  const float* x_q = (const float*)d_in[0];
  const float* x_kv = (const float*)d_in[1];
  const float* wq = (const float*)d_in[2];
  const float* bq = (const float*)d_in[3];
  const float* wk = (const float*)d_in[4];
  const float* bk = (const float*)d_in[5];
  const float* wv = (const float*)d_in[6];
  const float* bv = (const float*)d_in[7];

  float* out = (float*)d_out;
  float* attn_m_out = out + (size_t)BB * CHAN_ELEMS;  // tuple elem 2

  float* q = (float*)d_ws;
  float* kbuf = q + (size_t)BB * CHAN_ELEMS;
  float* vbuf = kbuf + (size_t)BB * CHAN_ELEMS;
  float* attn_ws = vbuf + (size_t)BB * CHAN_ELEMS;

  hipMemsetAsync(attn_ws, 0, (size_t)BB * CC * KK * sizeof(float), stream);

  const int total = BB * CC * SQ;
  dwconv3d_kernel<<<total / 256, 256, 0, stream>>>(x_q, wq, bq, q, total);
  dwconv3d_kernel<<<total / 256, 256, 0, stream>>>(x_kv, wk, bk, kbuf, total);
  dwconv3d_kernel<<<total / 256, 256, 0, stream>>>(x_kv, wv, bv, vbuf, total);

  attn_qk_kernel<<<dim3(64, BB), 512, 0, stream>>>(q, kbuf, attn_ws, SQ / 64);
  softmax_kernel<<<(BB * CC) / 8, 256, 0, stream>>>(attn_ws);
  topk_mask_kernel<<<(BB * KK) / 8, 256, 0, stream>>>(attn_ws, attn_m_out);
  attn_v_kernel<<<dim3(32, 4, BB), 256, 0, stream>>>(attn_m_out, vbuf, x_q,
                                                     out);
}